// HomoGraphEncoder_58815282151558
// MI455X (gfx1250) — compile-verified
//
#include <hip/hip_runtime.h>
#include <math.h>

// ---------------- problem constants (match reference) ----------------
#define NN   50000          // nodes (divisible by 16: 3125 * 16)
#define NE   400000         // edges
#define NTILES (NN / 16)    // 3125 row tiles
#define NL   8              // layers
#define NG   64             // graphs
#define ATT_SCALE 0.25f     // 1/sqrt(16)
#define LN_EPS 1e-5f
#define WAVES_PER_BLOCK 8

typedef __attribute__((ext_vector_type(2))) float v2f;
typedef __attribute__((ext_vector_type(8))) float v8f;

// ---------------- helpers ----------------
__device__ __forceinline__ void atomicMaxFloat(float* addr, float val) {
    int bits = __float_as_int(val);
    if (bits >= 0) {  // sign bit clear (val >= +0)
        atomicMax((int*)addr, bits);
    } else {          // negative (incl. -0): smaller uint bits == larger float
        atomicMin((unsigned int*)addr, (unsigned int)bits);
    }
}

// ---------------- WMMA GEMM: out[N x 64] = A[N x 64] @ W[64 x 64] + bias ----------------
// 256 threads = 8 waves per block. The 64x64 weight matrix (16 KB) and bias are
// staged in LDS once per block and shared by all 8 waves; each wave computes one
// 16x64 output tile with V_WMMA_F32_16X16X4_F32 (exact fp32), 16 K-steps of 4.
__global__ __launch_bounds__(256) void gemm64_wmma(const float* __restrict__ A,
                                                   const float* __restrict__ W,
                                                   const float* __restrict__ bias,
                                                   float* __restrict__ out,
                                                   int ntiles) {
    __shared__ float Wl[64 * 64];
    __shared__ float Bl[64];

    const int tid = threadIdx.x;

    // cooperative stage: 4096 floats = 1024 float4, 4 per thread
    {
        const float4* Wg4 = (const float4*)W;
        float4* Wl4 = (float4*)Wl;
#pragma unroll
        for (int i = 0; i < 4; ++i) Wl4[tid + 256 * i] = Wg4[tid + 256 * i];
        if (tid < 64) Bl[tid] = bias[tid];
    }
    __syncthreads();

    const int wave = tid >> 5;
    const int t    = tid & 31;
    const int hi   = t >> 4;       // lane half: 0 or 1
    const int lo   = t & 15;

    const long tile = (long)blockIdx.x * WAVES_PER_BLOCK + wave;
    if (tile >= ntiles) return;    // wave-uniform: EXEC stays all-ones inside WMMA
    const long base = tile * 16;

    v8f c[4] = {};                 // 4 N-tiles of 16 columns each

    // A operand: lane holds row (base+lo); lane half selects K offset {0,2} in each K4 chunk
    const float* arow = A + (base + lo) * 64 + 2 * hi;

#pragma unroll
    for (int kk = 0; kk < 16; ++kk) {
        const float2 av = *(const float2*)(arow + 4 * kk);
        v2f a; a[0] = av.x; a[1] = av.y;

        const int kr = 4 * kk + 2 * hi;              // K row pair for this lane half
        const float* w0 = Wl + kr * 64 + lo;         // B element (K=kr,   N=nt*16+lo)
        const float* w1 = w0 + 64;                   // B element (K=kr+1, N=nt*16+lo)
#pragma unroll
        for (int nt = 0; nt < 4; ++nt) {
            v2f b; b[0] = w0[nt * 16]; b[1] = w1[nt * 16];
            c[nt] = __builtin_amdgcn_wmma_f32_16x16x4_f32(
                false, a, false, b, (short)0, c[nt], false, false);
        }
    }

    // D layout: VGPR g holds M = g + 8*hi, N = nt*16 + lo
#pragma unroll
    for (int nt = 0; nt < 4; ++nt) {
        const int col = nt * 16 + lo;
        const float bb = Bl[col];
#pragma unroll
        for (int g = 0; g < 8; ++g) {
            out[(base + g + 8 * hi) * 64 + col] = c[nt][g] + bb;
        }
    }
}

// ---------------- fill ----------------
__global__ void fill_f32(float* __restrict__ p, float v, int n) {
    int i = blockIdx.x * blockDim.x + threadIdx.x;
    if (i < n) p[i] = v;
}

// ---------------- edge scoring: score[e,h] = (q[dst]·k[src])_h * scale; segment max ----------------
__global__ void edge_score_kernel(const float* __restrict__ q, const float* __restrict__ k,
                                  const int* __restrict__ src, const int* __restrict__ dst,
                                  float* __restrict__ score, float* __restrict__ mbuf) {
    int e = blockIdx.x * blockDim.x + threadIdx.x;
    if (e >= NE) return;
    const int sn = src[e], dn = dst[e];
    const float4* qp = (const float4*)(q + (long)dn * 64);
    const float4* kp = (const float4*)(k + (long)sn * 64);
    float acc[4] = {0.f, 0.f, 0.f, 0.f};
#pragma unroll
    for (int c4 = 0; c4 < 16; ++c4) {
        float4 a = qp[c4], b = kp[c4];
        acc[c4 >> 2] += a.x * b.x + a.y * b.y + a.z * b.z + a.w * b.w;
    }
#pragma unroll
    for (int hh = 0; hh < 4; ++hh) {
        float sc = acc[hh] * ATT_SCALE;
        score[(long)e * 4 + hh] = sc;
        atomicMaxFloat(&mbuf[(long)dn * 4 + hh], sc);
    }
}

// ---------------- fused exp + denom + value aggregation (unnormalized) ----------------
// alpha*v aggregated as (e*v); division by denom deferred to combine step.
__global__ void edge_exp_agg_kernel(const float* __restrict__ v,
                                    const int* __restrict__ src, const int* __restrict__ dst,
                                    const float* __restrict__ mbuf,
                                    const float* __restrict__ score,
                                    float* __restrict__ denom, float* __restrict__ agg) {
    long idx = (long)blockIdx.x * blockDim.x + threadIdx.x;
    if (idx >= (long)NE * 4) return;
    const int e  = (int)(idx >> 2);
    const int hh = (int)(idx & 3);
    const int sn = src[e], dn = dst[e];
    const float ex = __expf(score[idx] - mbuf[(long)dn * 4 + hh]);
    atomicAdd(&denom[(long)dn * 4 + hh], ex);
    const float4* vp = (const float4*)(v + (long)sn * 64 + hh * 16);
    float* ap = agg + (long)dn * 64 + hh * 16;
#pragma unroll
    for (int i2 = 0; i2 < 4; ++i2) {
        float4 vv = vp[i2];
        atomicAdd(ap + 4 * i2 + 0, ex * vv.x);
        atomicAdd(ap + 4 * i2 + 1, ex * vv.y);
        atomicAdd(ap + 4 * i2 + 2, ex * vv.z);
        atomicAdd(ap + 4 * i2 + 3, ex * vv.w);
    }
}

// ---------------- combine: h = LayerNorm(h + agg/denom + skip) ----------------
__global__ void combine_ln_kernel(float* __restrict__ h, const float* __restrict__ agg,
                                  const float* __restrict__ denom, const float* __restrict__ sb,
                                  const float* __restrict__ gamma, const float* __restrict__ beta) {
    int n = blockIdx.x * blockDim.x + threadIdx.x;
    if (n >= NN) return;
    float dn[4];
#pragma unroll
    for (int hh = 0; hh < 4; ++hh) dn[hh] = denom[(long)n * 4 + hh];

    float y[64];
    float mu = 0.f;
#pragma unroll
    for (int j = 0; j < 64; ++j) {
        float val = h[(long)n * 64 + j] + agg[(long)n * 64 + j] / dn[j >> 4] + sb[(long)n * 64 + j];
        y[j] = val;
        mu += val;
    }
    mu *= (1.f / 64.f);
    float var = 0.f;
#pragma unroll
    for (int j = 0; j < 64; ++j) {
        float d = y[j] - mu;
        var += d * d;
    }
    var *= (1.f / 64.f);
    const float r = rsqrtf(var + LN_EPS);
#pragma unroll
    for (int j = 0; j < 64; ++j) {
        h[(long)n * 64 + j] = (y[j] - mu) * r * gamma[j] + beta[j];
    }
}

// ---------------- graph pooling ----------------
__global__ void pool_accum_kernel(const float* __restrict__ hf, const int* __restrict__ batch,
                                  float* __restrict__ sum, float* __restrict__ cnt,
                                  float* __restrict__ mx) {
    int n = blockIdx.x * blockDim.x + threadIdx.x;
    if (n >= NN) return;
    const int g = batch[n];
    atomicAdd(&cnt[g], 1.0f);
#pragma unroll 4
    for (int j = 0; j < 64; ++j) {
        float val = hf[(long)n * 64 + j];
        atomicAdd(&sum[(long)g * 64 + j], val);
        atomicMaxFloat(&mx[(long)g * 64 + j], val);
    }
}

__global__ void pool_final_kernel(const float* __restrict__ sum, const float* __restrict__ cnt,
                                  const float* __restrict__ mx, float* __restrict__ out) {
    int idx = blockIdx.x * blockDim.x + threadIdx.x;
    if (idx >= NG * 64) return;
    const int g = idx >> 6;
    const int j = idx & 63;
    const float c = fmaxf(cnt[g], 1.0f);
    out[(long)g * 128 + j] = sum[idx] / c;
    out[(long)g * 128 + 64 + j] = mx[idx];
}

// ---------------- host launch ----------------
extern "C" void kernel_launch(void* const* d_in, const int* in_sizes, int n_in,
                              void* d_out, int out_size, void* d_ws, size_t ws_size,
                              hipStream_t stream) {
    (void)in_sizes; (void)n_in; (void)out_size; (void)ws_size;

    const float* x     = (const float*)d_in[0];
    const int*   ei    = (const int*)d_in[1];      // [2, E]
    const int*   batch = (const int*)d_in[2];      // [N]
    const float* W_in  = (const float*)d_in[3];
    const float* b_in  = (const float*)d_in[4];
    const float* Wq    = (const float*)d_in[5];
    const float* bq    = (const float*)d_in[6];
    const float* Wk    = (const float*)d_in[7];
    const float* bk    = (const float*)d_in[8];
    const float* Wv    = (const float*)d_in[9];
    const float* bv    = (const float*)d_in[10];
    const float* Ws    = (const float*)d_in[11];
    const float* bs    = (const float*)d_in[12];
    const float* gamma = (const float*)d_in[13];
    const float* beta  = (const float*)d_in[14];
    const float* W_out = (const float*)d_in[15];
    const float* b_out = (const float*)d_in[16];
    float* out = (float*)d_out;

    const int* src = ei;
    const int* dst = ei + NE;

    // workspace carve-out (floats): ~85 MB total, fits in global L2
    float* ws = (float*)d_ws;
    const size_t N64 = (size_t)NN * 64;
    float* hbuf  = ws;                              // [N,64] node features
    float* qb    = hbuf  + N64;                     // [N,64] (also reused for lin_out result)
    float* kb    = qb    + N64;
    float* vb    = kb    + N64;
    float* sb    = vb    + N64;
    float* agg   = sb    + N64;
    float* score = agg   + N64;                     // [E,4]
    float* mbuf  = score + (size_t)NE * 4;          // [N,4]
    float* denom = mbuf  + (size_t)NN * 4;          // [N,4]
    float* psum  = denom + (size_t)NN * 4;          // [G,64]
    float* pcnt  = psum  + (size_t)NG * 64;         // [G]
    float* pmax  = pcnt  + NG;                      // [G,64]

    const dim3 gemmGrid((NTILES + WAVES_PER_BLOCK - 1) / WAVES_PER_BLOCK);
    const dim3 gemmBlk(256);

    // h = x @ W_in + b_in
    gemm64_wmma<<<gemmGrid, gemmBlk, 0, stream>>>(x, W_in, b_in, hbuf, NTILES);

    for (int l = 0; l < NL; ++l) {
        const size_t wo = (size_t)l * 64 * 64;
        const size_t bo = (size_t)l * 64;
        gemm64_wmma<<<gemmGrid, gemmBlk, 0, stream>>>(hbuf, Wq + wo, bq + bo, qb, NTILES);
        gemm64_wmma<<<gemmGrid, gemmBlk, 0, stream>>>(hbuf, Wk + wo, bk + bo, kb, NTILES);
        gemm64_wmma<<<gemmGrid, gemmBlk, 0, stream>>>(hbuf, Wv + wo, bv + bo, vb, NTILES);
        gemm64_wmma<<<gemmGrid, gemmBlk, 0, stream>>>(hbuf, Ws + wo, bs + bo, sb, NTILES);

        fill_f32<<<(NN * 4 + 255) / 256, 256, 0, stream>>>(mbuf, -INFINITY, NN * 4);
        hipMemsetAsync(denom, 0, (size_t)NN * 4 * sizeof(float), stream);
        hipMemsetAsync(agg,   0, N64 * sizeof(float), stream);

        edge_score_kernel<<<(NE + 255) / 256, 256, 0, stream>>>(qb, kb, src, dst, score, mbuf);
        edge_exp_agg_kernel<<<(NE * 4 + 255) / 256, 256, 0, stream>>>(vb, src, dst, mbuf,
                                                                      score, denom, agg);
        combine_ln_kernel<<<(NN + 127) / 128, 128, 0, stream>>>(hbuf, agg, denom, sb,
                                                                gamma + bo, beta + bo);
    }

    // final projection (reuse qb) and pooling
    gemm64_wmma<<<gemmGrid, gemmBlk, 0, stream>>>(hbuf, W_out, b_out, qb, NTILES);

    hipMemsetAsync(psum, 0, ((size_t)NG * 64 + NG) * sizeof(float), stream); // psum + pcnt
    fill_f32<<<(NG * 64 + 255) / 256, 256, 0, stream>>>(pmax, -INFINITY, NG * 64);
    pool_accum_kernel<<<(NN + 255) / 256, 256, 0, stream>>>(qb, batch, psum, pcnt, pmax);
    pool_final_kernel<<<(NG * 64 + 255) / 256, 256, 0, stream>>>(psum, pcnt, pmax, out);
}